// CorrelatorK3_6502580486277
// MI455X (gfx1250) — compile-verified
//
#include <hip/hip_runtime.h>

// CDNA5 / gfx1250, wave32. Matrix math via V_WMMA_F32_16X16X4_F32; stage-2
// K-slice staging via the Tensor Data Mover (TENSOR_LOAD_TO_LDS + TENSORcnt).
typedef __attribute__((ext_vector_type(2))) float v2f;
typedef __attribute__((ext_vector_type(8))) float v8f;
typedef __attribute__((ext_vector_type(4))) unsigned int u32x4;
typedef __attribute__((ext_vector_type(4))) int i32x4;
typedef __attribute__((ext_vector_type(8))) int i32x8;

#define WMMA_F32_4(a, b, c) \
  __builtin_amdgcn_wmma_f32_16x16x4_f32(false, (a), false, (b), (short)0, (c), false, false)

#if defined(__has_builtin)
#if __has_builtin(__builtin_amdgcn_tensor_load_to_lds) && \
    __has_builtin(__builtin_amdgcn_s_wait_tensorcnt)
#define CORR3_TDM 1
#endif
#endif

constexpr int   kB = 8, kN = 256, kD = 64, kF = 64;
constexpr long  kM = (long)kB * kN * kN;       // 524288 rows of rbf_*
constexpr long  kK2 = (long)kN * kF;           // 16384 contraction length, stage 2
constexpr float kInterval = 0.02f;

// ---------------------------------------------------------------------------
// Stage 1: four 64x64 projections + elementwise products (unchanged).
//   Aout[b][n][i][f] = (rbf0·Q0w + Q0b) * (rbfd·Qw + Qb)     (row-major by m)
//   Bout[b][j][i][f] = (rbf0·R0w + R0b) * (rbfd·Rw + Rb)     (j/i transposed!)
// ---------------------------------------------------------------------------
constexpr int S1_ROWS = 128;
constexpr int XP = 68;   // stride mod 64 = 4 -> conflict-free A frags
constexpr int FP = 80;   // transposed weights [d][f]: 2*80 mod 64 = 32 -> conflict-free B frags

__global__ __launch_bounds__(256)
void corr3_stage1(const float* __restrict__ rbf0, const float* __restrict__ rbfd,
                  const float* __restrict__ Q0w, const float* __restrict__ Q0b,
                  const float* __restrict__ Qw,  const float* __restrict__ Qb,
                  const float* __restrict__ R0w, const float* __restrict__ R0b,
                  const float* __restrict__ Rw,  const float* __restrict__ Rb,
                  float* __restrict__ Aout, float* __restrict__ Bout)
{
  __shared__ float lx0[S1_ROWS * XP];
  __shared__ float lxd[S1_ROWS * XP];
  __shared__ float w0[kD * FP];
  __shared__ float w1[kD * FP];
  __shared__ float w2[kD * FP];
  __shared__ float w3[kD * FP];
  __shared__ float bia[4][kF];

  const int  t    = threadIdx.x;
  const int  lane = t & 31;
  const int  wave = t >> 5;
  const long mBase = (long)blockIdx.x * S1_ROWS;

  for (int e = t; e < kF * kD; e += 256) {
    const int f = e >> 6, d = e & 63;
    w0[d * FP + f] = Q0w[e];
    w1[d * FP + f] = Qw[e];
    w2[d * FP + f] = R0w[e];
    w3[d * FP + f] = Rw[e];
  }
  if (t < kF) {
    bia[0][t] = Q0b[t]; bia[1][t] = Qb[t]; bia[2][t] = R0b[t]; bia[3][t] = Rb[t];
  }
  for (int e = t; e < S1_ROWS * 16; e += 256) {
    const int  r = e >> 4, c4 = e & 15;
    const long src = (mBase + r) * 16 + c4;
    const float4 a  = ((const float4*)rbf0)[src];
    const float4 dd = ((const float4*)rbfd)[src];
    *(float4*)&lx0[r * XP + c4 * 4] = a;
    *(float4*)&lxd[r * XP + c4 * 4] = dd;
  }
  __syncthreads();

  const int lr    = lane & 15;
  const int koff  = (lane >> 4) << 1;
  const int vroff = (lane >> 4) << 3;
  const int rowT  = wave << 4;
  const float* paRow0 = &lx0[(rowT + lr) * XP + koff];
  const float* paRowD = &lxd[(rowT + lr) * XP + koff];

  for (int fb = 0; fb < 4; ++fb) {
    const int fcol = (fb << 4) + lr;
    v8f cQ0 = {}, cQ = {}, cR0 = {}, cR = {};
#pragma unroll
    for (int kk = 0; kk < kD; kk += 4) {
      const v2f a0 = *(const v2f*)(paRow0 + kk);
      const v2f ad = *(const v2f*)(paRowD + kk);
      const int wi = (kk + koff) * FP + fcol;
      v2f b0, b1, b2, b3;
      b0.x = w0[wi]; b0.y = w0[wi + FP];
      b1.x = w1[wi]; b1.y = w1[wi + FP];
      b2.x = w2[wi]; b2.y = w2[wi + FP];
      b3.x = w3[wi]; b3.y = w3[wi + FP];
      cQ0 = WMMA_F32_4(a0, b0, cQ0);
      cQ  = WMMA_F32_4(ad, b1, cQ);
      cR0 = WMMA_F32_4(a0, b2, cR0);
      cR  = WMMA_F32_4(ad, b3, cR);
    }
    const float bq0 = bia[0][fcol], bq = bia[1][fcol];
    const float br0 = bia[2][fcol], br = bia[3][fcol];
#pragma unroll
    for (int v = 0; v < 8; ++v) {
      const long  g  = mBase + rowT + vroff + v;
      const float qv = (cQ0[v] + bq0) * (cQ[v] + bq);
      const float rv = (cR0[v] + br0) * (cR[v] + br);
      Aout[g * kF + fcol] = qv;
      const int bi = (int)(g >> 16);
      const int ii = ((int)g >> 8) & 255;
      const int jj = (int)g & 255;
      Bout[((long)(bi * kN + jj) * kN + ii) * kF + fcol] = rv;
    }
  }
}

// ---------------------------------------------------------------------------
// TDM helper: 2D tile DMA, 64 rows x 64 f32, row pitch 16384 f32 in memory,
// LDS destination padded by hardware: 64 DWORDs then +4 DWORDs -> pitch 68.
// D# bitfields per CDNA5 ISA §8.3-8.4 (group0: count/lds/global/type,
// group1: data_size, pad, tensor dims, tile dims, stride).
// ---------------------------------------------------------------------------
#ifdef CORR3_TDM
__device__ __forceinline__ void tdm_load_tile(const float* gsrc, unsigned lds_off) {
  const unsigned long long ga = (unsigned long long)(const void*)gsrc;
  u32x4 g0;
  g0[0] = 1u;                                            // count=1, user mode
  g0[1] = lds_off;                                       // lds_addr (bytes)
  g0[2] = (unsigned)ga;                                  // global_addr[31:0]
  g0[3] = (unsigned)((ga >> 32) & 0x01FFFFFFull)         // global_addr[56:32]
        | 0x80000000u;                                   // type=2 ("image")
  i32x8 g1;
  g1[0] = (2 << 16)      // data_size = 4B
        | (1 << 20)      // pad_enable
        | (5 << 22)      // pad_interval: 64 DWORDs
        | (3 << 25);     // pad_amount:   4 DWORDs  -> LDS pitch 68 floats
  g1[1] = (int)(16384u << 16);   // tensor_dim0 = 16384 (low16 in [31:16])
  g1[2] = (int)(256u << 16);     // tensor_dim0 hi=0 | tensor_dim1 = 256
  g1[3] = (int)(64u << 16);      // tensor_dim1 hi=0 | tile_dim0 = 64
  g1[4] = 64;                    // tile_dim1 = 64, tile_dim2 = 0 (2D)
  g1[5] = 16384;                 // tensor_dim0_stride[31:0]
  g1[6] = 0;                     // stride hi | tensor_dim1_stride lo
  g1[7] = 0;
  const i32x4 gz = {0, 0, 0, 0};
#if __clang_major__ >= 23
  const i32x8 gz8 = {0, 0, 0, 0, 0, 0, 0, 0};
  __builtin_amdgcn_tensor_load_to_lds(g0, g1, gz, gz, gz8, 0);
#else
  __builtin_amdgcn_tensor_load_to_lds(g0, g1, gz, gz, 0);
#endif
}
#endif

// ---------------------------------------------------------------------------
// Stage 2: per-batch NT GEMM  C[b] (256x256) = A_b (256x16384) x Bt_b^T, *0.02.
// 64x64 tile per 128-thread workgroup; wave w owns a 32x32 quadrant as a
// 2x2 grid of 16x16 WMMA tiles (shared A/B fragments -> 1 ds_b64 per wmma).
// Double-buffered 64-wide K slices staged by the TDM (wave 0 issues, all
// waves consume after s_wait_tensorcnt + barrier).
// ---------------------------------------------------------------------------
constexpr int KP = 68;

__global__ __launch_bounds__(128)
void corr3_stage2(const float* __restrict__ A, const float* __restrict__ Bt,
                  float* __restrict__ out)
{
  __shared__ float sA[2][64 * KP];
  __shared__ float sB[2][64 * KP];

  const int t    = threadIdx.x;
  const int lane = t & 31;
  const int wave = t >> 5;
  const int b     = blockIdx.z;
  const int nBase = blockIdx.y * 64;
  const int jBase = blockIdx.x * 64;

  const float* Ab = A  + ((long)b * kN + nBase) * kK2;   // [64 rows][16384]
  const float* Bb = Bt + ((long)b * kN + jBase) * kK2;   // [64 rows][16384]

  const int rA0 = (wave >> 1) << 5;    // 0 | 32  (n quadrant)
  const int cB0 = (wave & 1) << 5;     // 0 | 32  (j quadrant)
  const int lr    = lane & 15;
  const int koff  = (lane >> 4) << 1;
  const int vroff = (lane >> 4) << 3;

  constexpr int STAGES = (int)(kK2 / 64);      // 256
  v8f c00 = {}, c01 = {}, c10 = {}, c11 = {};

#ifdef CORR3_TDM
  const unsigned ldsA0 = (unsigned)(unsigned long long)&sA[0][0];
  const unsigned ldsB0 = (unsigned)(unsigned long long)&sB[0][0];
  constexpr unsigned BUFB = 64u * KP * 4u;     // bytes per buffer

  if (wave == 0) {                             // prime stage 0 (2 tensor ops)
    tdm_load_tile(Ab, ldsA0);
    tdm_load_tile(Bb, ldsB0);
  }
  for (int s = 0; s < STAGES; ++s) {
    const int buf = s & 1;
    if (wave == 0) {
      if (s + 1 < STAGES) {                    // prefetch next slice
        const int nb = (s + 1) & 1;
        tdm_load_tile(Ab + (long)(s + 1) * 64, ldsA0 + nb * BUFB);
        tdm_load_tile(Bb + (long)(s + 1) * 64, ldsB0 + nb * BUFB);
        __builtin_amdgcn_s_wait_tensorcnt(2);  // stage-s pair has landed
      } else {
        __builtin_amdgcn_s_wait_tensorcnt(0);
      }
    }
    __syncthreads();                           // publish slice s to all waves
    const float* pa0 = &sA[buf][(rA0 + lr) * KP + koff];
    const float* pa1 = pa0 + 16 * KP;
    const float* pb0 = &sB[buf][(cB0 + lr) * KP + koff];
    const float* pb1 = pb0 + 16 * KP;
#pragma unroll
    for (int kk = 0; kk < 64; kk += 4) {
      const v2f a0 = *(const v2f*)(pa0 + kk);
      const v2f a1 = *(const v2f*)(pa1 + kk);
      const v2f b0 = *(const v2f*)(pb0 + kk);
      const v2f b1 = *(const v2f*)(pb1 + kk);
      c00 = WMMA_F32_4(a0, b0, c00);
      c01 = WMMA_F32_4(a0, b1, c01);
      c10 = WMMA_F32_4(a1, b0, c10);
      c11 = WMMA_F32_4(a1, b1, c11);
    }
    __syncthreads();                           // all done with buf before next issue refills it
  }
#else
  // Fallback path (no TDM builtin): plain coalesced copy each stage.
  for (int s = 0; s < STAGES; ++s) {
    __syncthreads();
    const long kS = (long)s * 64;
    for (int e = t; e < 64 * 16; e += 128) {
      const int rr = e >> 4, c4 = e & 15;
      *(float4*)&sA[0][rr * KP + c4 * 4] = *(const float4*)(Ab + rr * kK2 + kS + c4 * 4);
      *(float4*)&sB[0][rr * KP + c4 * 4] = *(const float4*)(Bb + rr * kK2 + kS + c4 * 4);
    }
    __syncthreads();
    const float* pa0 = &sA[0][(rA0 + lr) * KP + koff];
    const float* pa1 = pa0 + 16 * KP;
    const float* pb0 = &sB[0][(cB0 + lr) * KP + koff];
    const float* pb1 = pb0 + 16 * KP;
#pragma unroll
    for (int kk = 0; kk < 64; kk += 4) {
      const v2f a0 = *(const v2f*)(pa0 + kk);
      const v2f a1 = *(const v2f*)(pa1 + kk);
      const v2f b0 = *(const v2f*)(pb0 + kk);
      const v2f b1 = *(const v2f*)(pb1 + kk);
      c00 = WMMA_F32_4(a0, b0, c00);
      c01 = WMMA_F32_4(a0, b1, c01);
      c10 = WMMA_F32_4(a1, b0, c10);
      c11 = WMMA_F32_4(a1, b1, c11);
    }
  }
#endif

  float* ob = out + (long)b * kN * kN;
  const int n0 = nBase + rA0 + vroff;
  const int j0 = jBase + cB0 + lr;
#pragma unroll
  for (int v = 0; v < 8; ++v) {
    ob[(n0 + v) * kN + j0]           = c00[v] * kInterval;
    ob[(n0 + v) * kN + j0 + 16]      = c01[v] * kInterval;
    ob[(n0 + 16 + v) * kN + j0]      = c10[v] * kInterval;
    ob[(n0 + 16 + v) * kN + j0 + 16] = c11[v] * kInterval;
  }
}

// ---------------------------------------------------------------------------
extern "C" void kernel_launch(void* const* d_in, const int* in_sizes, int n_in,
                              void* d_out, int out_size, void* d_ws, size_t ws_size,
                              hipStream_t stream) {
  const float* rbf0 = (const float*)d_in[0];
  const float* rbfd = (const float*)d_in[1];
  const float* Q0w  = (const float*)d_in[2];
  const float* Q0b  = (const float*)d_in[3];
  const float* Qw   = (const float*)d_in[4];
  const float* Qb   = (const float*)d_in[5];
  const float* R0w  = (const float*)d_in[6];
  const float* R0b  = (const float*)d_in[7];
  const float* Rw   = (const float*)d_in[8];
  const float* Rb   = (const float*)d_in[9];

  float* Aw = (float*)d_ws;                     // 524288 x 64 f32 = 128 MiB
  float* Bw = Aw + (size_t)kM * kF;             // 128 MiB more

  corr3_stage1<<<(int)(kM / S1_ROWS), 256, 0, stream>>>(
      rbf0, rbfd, Q0w, Q0b, Qw, Qb, R0w, R0b, Rw, Rb, Aw, Bw);

  corr3_stage2<<<dim3(kN / 64, kN / 64, kB), 128, 0, stream>>>(
      Aw, Bw, (float*)d_out);
}